// DGMM_40621800686202
// MI455X (gfx1250) — compile-verified
//
#include <hip/hip_runtime.h>
#include <math.h>

#define NPTS 65536
#define KMIX 16
#define DDIM 128
#define LOG_2PI 1.8378770664093453f

typedef __attribute__((ext_vector_type(2))) float v2f;
typedef __attribute__((ext_vector_type(8))) float v8f;

// async global->LDS copy of 16 bytes (gfx1250), tracked by ASYNCcnt
#define ASYNC_CP16(ldsoff_u32, gsrc_ptr)                                   \
    asm volatile("global_load_async_to_lds_b128 %0, %1, off"               \
                 :: "v"(ldsoff_u32), "v"(gsrc_ptr) : "memory")

// ---- workspace layout (float offsets) ----
#define OFF_NK      0            // 16
#define OFF_S1      16           // 16*128
#define OFF_S2      2064         // 16*128*128
#define OFF_HLD     264208       // 16  (sum of log diag(L) per k)
#define OFF_CQ      264224       // 16  (mu^T Cinv mu per k)
#define OFF_EACC    264240       // 1
#define OFF_CDACC   264241       // 1
#define ZERO_COUNT  264242
#define OFF_MU      264256       // 16*128
#define OFF_CINV    266304       // 16*128*128
#define OFF_X       528448       // 16*128*128 (L^-1 per k)
#define OFF_B       790592       // 16*128
#define OFF_KC      792640       // 16  (per-k log constant)
#define OFF_S       792656       // N*K  (-0.5*q2 + z.b)

// ---------------- kernel 0: zero accumulators ----------------
__global__ void k_zero(float* ws) {
    int i = blockIdx.x * 256 + threadIdx.x;
    if (i < ZERO_COUNT) ws[i] = 0.0f;
}

// ---------------- kernel 1: Nk and S1 = gamma^T z ----------------
// grid 64 blocks x 128 threads, each block handles 1024 rows
__global__ void k_moment1(const float* __restrict__ z,
                          const float* __restrict__ gamma,
                          float* __restrict__ ws) {
    __shared__ float g[128];      // 8 rows x 16 k
    __shared__ float sNk[16];
    const int tid = threadIdx.x;
    const int nb = blockIdx.x * 1024;
    float acc[KMIX];
#pragma unroll
    for (int k = 0; k < KMIX; ++k) acc[k] = 0.0f;
    float accNk = 0.0f;
    const int kk = tid & 15;
    const int rsub = tid >> 4;   // 0..7

    for (int r0 = 0; r0 < 1024; r0 += 8) {
        __syncthreads();
        float gv = gamma[(nb + r0 + rsub) * KMIX + kk];
        g[rsub * 16 + kk] = gv;
        accNk += gv;
        __syncthreads();
#pragma unroll
        for (int rr = 0; rr < 8; ++rr) {
            float zv = z[(size_t)(nb + r0 + rr) * DDIM + tid];
#pragma unroll
            for (int k = 0; k < KMIX; ++k) acc[k] += g[rr * 16 + k] * zv;
        }
    }
    if (tid < 16) sNk[tid] = 0.0f;
    __syncthreads();
    atomicAdd(&sNk[kk], accNk);
    __syncthreads();
    if (tid < 16) atomicAdd(&ws[OFF_NK + tid], sNk[tid]);
#pragma unroll
    for (int k = 0; k < KMIX; ++k)
        atomicAdd(&ws[OFF_S1 + k * DDIM + tid], acc[k]);
}

// ---------------- kernel 2: S2_k += sum_n gamma[n,k] z z^T (WMMA f32) ----
// grid (32 n-chunks, 8 k-pairs), block 256 = 8 waves; chunk = 2048 rows.
// Two mixture components per block share the LDS z tile (B operand reuse).
// Double-buffered async global->LDS staging pipelined behind WMMA compute.
// launch_bounds(256,2): 8 waves/WGP, up to 512 VGPRs/wave -> no acc spills.
__global__ void __launch_bounds__(256, 2)
k_s2(const float* __restrict__ z,
     const float* __restrict__ gamma,
     float* __restrict__ ws) {
    __shared__ __align__(16) float zt[2][32 * 132];
    __shared__ float gt[2][2][32];
    const int tid  = threadIdx.x;
    const int lane = tid & 31;
    const int w    = tid >> 5;
    const int half = lane >> 4;
    const int l16  = lane & 15;
    const int k0   = blockIdx.y * 2;
    const int cbase = blockIdx.x * 2048;
    float* S2 = ws + OFF_S2;

    v8f acc0[8], acc1[8];
#pragma unroll
    for (int j = 0; j < 8; ++j) { acc0[j] = (v8f)0.0f; acc1[j] = (v8f)0.0f; }

    auto issue_tile = [&](int nb, int buf) {
#pragma unroll
        for (int p = 0; p < 4; ++p) {
            int f = (p * 256 + tid) * 4;           // 0..4095
            int row = f >> 7, col = f & 127;
            const float* src = z + (size_t)(nb + row) * DDIM + col;
            unsigned dst = (unsigned)(size_t)&zt[buf][row * 132 + col];
            ASYNC_CP16(dst, src);
        }
        if (tid < 64)
            gt[buf][tid >> 5][tid & 31] =
                gamma[(size_t)(nb + (tid & 31)) * KMIX + k0 + (tid >> 5)];
    };

    issue_tile(cbase, 0);
    for (int it = 0; it < 64; ++it) {
        const int buf = it & 1;
        if (it < 63) {
            issue_tile(cbase + (it + 1) * 32, buf ^ 1);
            asm volatile("s_wait_asynccnt 0x4" ::: "memory");
        } else {
            asm volatile("s_wait_asynccnt 0x0" ::: "memory");
        }
        __syncthreads();   // tile `buf` visible to all waves

        const float* zb = &zt[buf][0];
        const float* g0 = &gt[buf][0][0];
        const float* g1 = &gt[buf][1][0];
        const int d = w * 16 + l16;
#pragma unroll
        for (int s4 = 0; s4 < 8; ++s4) {
            const int r0 = s4 * 4 + 2 * half;
            float za = zb[r0 * 132 + d];
            float zc = zb[(r0 + 1) * 132 + d];
            v2f a0, a1;
            a0.x = g0[r0] * za;  a0.y = g0[r0 + 1] * zc;
            a1.x = g1[r0] * za;  a1.y = g1[r0 + 1] * zc;
#pragma unroll
            for (int j = 0; j < 8; ++j) {
                const int col = j * 16 + l16;
                v2f b;
                b.x = zb[r0 * 132 + col];
                b.y = zb[(r0 + 1) * 132 + col];
                acc0[j] = __builtin_amdgcn_wmma_f32_16x16x4_f32(
                    false, a0, false, b, (short)0, acc0[j], false, false);
                acc1[j] = __builtin_amdgcn_wmma_f32_16x16x4_f32(
                    false, a1, false, b, (short)0, acc1[j], false, false);
            }
        }
        __syncthreads();   // all waves done reading `buf` before it is refilled
    }
    // C layout: VGPR r -> row r + 8*half, lane -> column l16
#pragma unroll
    for (int j = 0; j < 8; ++j) {
#pragma unroll
        for (int r = 0; r < 8; ++r) {
            int drow = w * 16 + r + 8 * half;
            int dcol = j * 16 + l16;
            atomicAdd(&S2[((size_t)k0 * DDIM + drow) * DDIM + dcol], acc0[j][r]);
            atomicAdd(&S2[((size_t)(k0 + 1) * DDIM + drow) * DDIM + dcol], acc1[j][r]);
        }
    }
}

// ---------------- kernel 3: per-k cov, Cholesky, inverse, constants -----
// grid 16 blocks x 128 threads
__global__ void k_finalize(float* __restrict__ ws) {
    __shared__ float A[DDIM * DDIM];   // 64KB: cov -> L (lower)
    const int tid = threadIdx.x;
    const int k = blockIdx.x;
    float* muG = ws + OFF_MU + k * DDIM;
    float* Xg  = ws + OFF_X  + (size_t)k * DDIM * DDIM;
    float* Cinv = ws + OFF_CINV + (size_t)k * DDIM * DDIM;
    const float Nkk = ws[OFF_NK + k];
    const float invNk = 1.0f / Nkk;

    const float mu_i = ws[OFF_S1 + k * DDIM + tid] * invNk;
    muG[tid] = mu_i;
    __threadfence();
    __syncthreads();

    const float* S2 = ws + OFF_S2 + (size_t)k * DDIM * DDIM;
    for (int jj = 0; jj < DDIM; ++jj)
        A[tid * DDIM + jj] = S2[tid * DDIM + jj] * invNk - mu_i * muG[jj];
    A[tid * DDIM + tid] += 1e-20f;
    __syncthreads();

    // cov_diag contribution (before Cholesky overwrites diag)
    atomicAdd(&ws[OFF_CDACC], 1.0f / A[tid * DDIM + tid]);

    // in-place lower Cholesky
    for (int j = 0; j < DDIM; ++j) {
        __syncthreads();
        float ajj = sqrtf(A[j * DDIM + j]);
        float lij = 0.0f;
        if (tid >= j) lij = A[tid * DDIM + j] / ajj;
        __syncthreads();
        if (tid >= j) A[tid * DDIM + j] = lij;
        __syncthreads();
        if (tid > j) {
            float lj = A[tid * DDIM + j];
            for (int c = j + 1; c <= tid; ++c)
                A[tid * DDIM + c] -= lj * A[c * DDIM + j];
        }
    }
    __syncthreads();
    atomicAdd(&ws[OFF_HLD + k], logf(A[tid * DDIM + tid]));

    // forward solve: column tid of X = L^{-1}
    {
        float xv[DDIM];
        const int c = tid;
        for (int i = c; i < DDIM; ++i) {
            float s = 0.0f;
            for (int j = c; j < i; ++j) s += A[i * DDIM + j] * xv[j];
            xv[i] = (((i == c) ? 1.0f : 0.0f) - s) / A[i * DDIM + i];
        }
        for (int i = c; i < DDIM; ++i) Xg[i * DDIM + c] = xv[i];
    }
    __threadfence();
    __syncthreads();

    // Cinv = X^T X  (column tid)
    for (int i = 0; i < DDIM; ++i) {
        int m0 = (i > tid) ? i : tid;
        float s = 0.0f;
        for (int m = m0; m < DDIM; ++m)
            s += Xg[m * DDIM + i] * Xg[m * DDIM + tid];
        Cinv[i * DDIM + tid] = s;
    }
    __threadfence();
    __syncthreads();

    // b = Cinv mu,  c_k = mu^T b
    {
        float s = 0.0f;
        for (int jj = 0; jj < DDIM; ++jj)
            s += Cinv[tid * DDIM + jj] * muG[jj];
        ws[OFF_B + k * DDIM + tid] = s;
        atomicAdd(&ws[OFF_CQ + k], s * mu_i);
    }
    __threadfence();
    __syncthreads();

    if (tid == 0) {
        // log phi - 32*log(2pi) - 0.5*sum(log Ljj) - 0.5*c_k
        ws[OFF_KC + k] = logf(Nkk / (float)NPTS) - 32.0f * LOG_2PI
                         - 0.5f * ws[OFF_HLD + k] - 0.5f * ws[OFF_CQ + k];
    }
}

// ---------------- kernel 4: s[n,k] = -0.5 z^T Cinv z + z.b (WMMA) -------
// grid (N/128, K), block 256 = 8 waves; wave w owns 16 rows of n.
// Cinv staged to LDS with async copies.
__global__ void __launch_bounds__(256, 2)
k_quad(const float* __restrict__ z,
       float* __restrict__ ws) {
    __shared__ __align__(16) float Ci[DDIM * DDIM];  // 64KB
    const int tid  = threadIdx.x;
    const int lane = tid & 31;
    const int w    = tid >> 5;
    const int half = lane >> 4;
    const int l16  = lane & 15;
    const int k    = blockIdx.y;
    const int nbase = blockIdx.x * 128;
    const float* CinvG = ws + OFF_CINV + (size_t)k * DDIM * DDIM;
    const float* bv = ws + OFF_B + k * DDIM;
    float* sG = ws + OFF_S;

#pragma unroll
    for (int p = 0; p < 16; ++p) {
        int f = (p * 256 + tid) * 4;
        const float* src = CinvG + f;
        unsigned dst = (unsigned)(size_t)&Ci[f];
        ASYNC_CP16(dst, src);
    }
    asm volatile("s_wait_asynccnt 0x0" ::: "memory");
    __syncthreads();

    const float* zr = z + (size_t)(nbase + 16 * w + l16) * DDIM;
    v8f acc[8];
#pragma unroll
    for (int j = 0; j < 8; ++j) acc[j] = (v8f)0.0f;

    for (int t = 0; t < 32; ++t) {
        v2f a = *(const v2f*)(zr + 4 * t + 2 * half);
        const int r0 = 4 * t + 2 * half;
#pragma unroll
        for (int j = 0; j < 8; ++j) {
            const int col = j * 16 + l16;
            v2f b;
            b.x = Ci[r0 * DDIM + col];
            b.y = Ci[(r0 + 1) * DDIM + col];
            acc[j] = __builtin_amdgcn_wmma_f32_16x16x4_f32(
                false, a, false, b, (short)0, acc[j], false, false);
        }
    }

    // fused row-dot: s_row = sum_col z*(b - 0.5*Y)
    float p[8];
#pragma unroll
    for (int r = 0; r < 8; ++r) p[r] = 0.0f;
    const float* zb = z + (size_t)(nbase + 16 * w + 8 * half) * DDIM;
#pragma unroll
    for (int j = 0; j < 8; ++j) {
        const int col = j * 16 + l16;
        float bj = bv[col];
#pragma unroll
        for (int r = 0; r < 8; ++r) {
            float zv = zb[r * DDIM + col];
            p[r] += zv * (bj - 0.5f * acc[j][r]);
        }
    }
#pragma unroll
    for (int r = 0; r < 8; ++r) {
        p[r] += __shfl_xor(p[r], 1, 16);
        p[r] += __shfl_xor(p[r], 2, 16);
        p[r] += __shfl_xor(p[r], 4, 16);
        p[r] += __shfl_xor(p[r], 8, 16);
    }
#pragma unroll
    for (int r = 0; r < 8; ++r)
        if (l16 == r)
            sG[(size_t)(nbase + 16 * w + r + 8 * half) * KMIX + k] = p[r];
}

// ---------------- kernel 5: per-n energy + reduce ----------------
__global__ void k_energy(float* __restrict__ ws) {
    __shared__ float red[256];
    const int tid = threadIdx.x;
    const int n = blockIdx.x * 256 + tid;
    const float* sG = ws + OFF_S;
    float mix = 0.0f;
#pragma unroll
    for (int k = 0; k < KMIX; ++k)
        mix += expf(ws[OFF_KC + k] + sG[(size_t)n * KMIX + k]);
    red[tid] = -logf(mix + 1e-20f);
    __syncthreads();
    for (int s = 128; s > 0; s >>= 1) {
        if (tid < s) red[tid] += red[tid + s];
        __syncthreads();
    }
    if (tid == 0) atomicAdd(&ws[OFF_EACC], red[0]);
}

// ---------------- kernel 6: final scalar ----------------
__global__ void k_final(const float* __restrict__ ws, float* __restrict__ out) {
    out[0] = ws[OFF_EACC] * (1.0f / (float)NPTS) + 0.005f * ws[OFF_CDACC];
}

extern "C" void kernel_launch(void* const* d_in, const int* in_sizes, int n_in,
                              void* d_out, int out_size, void* d_ws, size_t ws_size,
                              hipStream_t stream) {
    (void)in_sizes; (void)n_in; (void)out_size; (void)ws_size;
    const float* z = (const float*)d_in[0];
    const float* gamma = (const float*)d_in[1];
    float* ws = (float*)d_ws;
    float* out = (float*)d_out;

    k_zero<<<(ZERO_COUNT + 255) / 256, 256, 0, stream>>>(ws);
    k_moment1<<<64, 128, 0, stream>>>(z, gamma, ws);
    k_s2<<<dim3(32, 8), 256, 0, stream>>>(z, gamma, ws);
    k_finalize<<<16, 128, 0, stream>>>(ws);
    k_quad<<<dim3(NPTS / 128, 16), 256, 0, stream>>>(z, ws);
    k_energy<<<NPTS / 256, 256, 0, stream>>>(ws);
    k_final<<<1, 32, 0, stream>>>(ws, out);
}